// ElasticPooling_15358803050680
// MI455X (gfx1250) — compile-verified
//
#include <hip/hip_runtime.h>

// ElasticPooling: x (16, 1024, 32, 128) f32 -> out (16, 128, 16, 128) f32
// out[b,p1,wv,c] = max_{r<8, j<2} x[b, p1*8+r, wv*2+j, c]
//
// Pure streaming reduction: 256 MB in + 16 MB out, ~15 fmax/output.
// HBM-bound: 272 MB / 23.3 TB/s ~= 11.7 us on MI455X. Strategy:
//  - one float4 of output per thread (lanes cover c -> 512B contiguous/wave/load)
//  - 16 independent global_load_b128 with NT temporal hint (stream-once data,
//    input > 192MB L2), then tree-reduce, NT b128 store.

typedef float v4f __attribute__((ext_vector_type(4)));

__device__ __forceinline__ v4f vmax4(v4f a, v4f b) {
    v4f r;
    r.x = __builtin_fmaxf(a.x, b.x);
    r.y = __builtin_fmaxf(a.y, b.y);
    r.z = __builtin_fmaxf(a.z, b.z);
    r.w = __builtin_fmaxf(a.w, b.w);
    return r;
}

// Shapes (in float4 units along C): C4 = 128/4 = 32
//   input  float4 index: ((bp*8 + r)*32 + (wv*2 + j))*32 + c4,  bp = b*128 + p1
//   output float4 index: tid = ((bp)*16 + wv)*32 + c4
__global__ __launch_bounds__(256) void elastic_pool_max_kernel(
    const v4f* __restrict__ x, v4f* __restrict__ out)
{
    const int tid = blockIdx.x * 256 + threadIdx.x;   // 0 .. 1,048,575
    const int c4  = tid & 31;          // float4 slot within C
    const int wv  = (tid >> 5) & 15;   // word-vector bin
    const int bp  = tid >> 9;          // b*128 + p1  (0 .. 2047)

    // base points at (h = bp*8, w = wv*2, c4); r stride = 32*32 = 1024 f4,
    // j stride = 32 f4.
    const v4f* __restrict__ base =
        x + ((size_t)bp * 8 * 32 + (size_t)wv * 2) * 32 + c4;

    v4f v[16];
#pragma unroll
    for (int r = 0; r < 8; ++r) {
        v[2 * r]     = __builtin_nontemporal_load(base + (size_t)r * 1024);
        v[2 * r + 1] = __builtin_nontemporal_load(base + (size_t)r * 1024 + 32);
    }

    // Tree reduction: 15 vector-max (compiler may fuse into v_max3_f32).
#pragma unroll
    for (int s = 8; s > 0; s >>= 1) {
#pragma unroll
        for (int i = 0; i < s; ++i) {
            v[i] = vmax4(v[i], v[i + s]);
        }
    }

    __builtin_nontemporal_store(v[0], out + tid);
}

extern "C" void kernel_launch(void* const* d_in, const int* in_sizes, int n_in,
                              void* d_out, int out_size, void* d_ws, size_t ws_size,
                              hipStream_t stream) {
    (void)in_sizes; (void)n_in; (void)d_ws; (void)ws_size;
    const v4f* x  = (const v4f*)d_in[0];
    v4f* out      = (v4f*)d_out;

    const int n_out_f4 = out_size / 4;          // 4,194,304 / 4 = 1,048,576
    const int threads  = 256;
    const int blocks   = n_out_f4 / threads;    // 4096

    elastic_pool_max_kernel<<<blocks, threads, 0, stream>>>(x, out);
}